// k1_GNN_87729001988943
// MI455X (gfx1250) — compile-verified
//
#include <hip/hip_runtime.h>
#include <math.h>

// ----------------------------------------------------------------------------
// NNConv GNN forward for MI455X (gfx1250, wave32).
//   msg[e,o] = sum_{k,i} h[e,k] * xs[e,i] * w2[k, i*mo+o]  (+ b2 term)
// realized as on-the-fly P = (h x xs) outer-product fragments fed to
// V_WMMA_F32_16X16X4_F32.  h/xs tiles are staged into LDS by the Tensor Data
// Mover (tensor_load_to_lds) with hardware pad-insertion (+1 dword per row)
// to avoid LDS bank conflicts.  w2 is pre-packed (layer 0 only) so the hot
// loop is guard-free and fully unrolled per K-chunk.
// ----------------------------------------------------------------------------

typedef __attribute__((ext_vector_type(2))) float v2f;
typedef __attribute__((ext_vector_type(8))) float v8f;
typedef __attribute__((ext_vector_type(4))) unsigned v4u;
typedef __attribute__((ext_vector_type(8))) unsigned v8u;

static __device__ __forceinline__ float eluf(float v) {
    return v > 0.0f ? v : (expf(v) - 1.0f);
}

// generic -> LDS byte offset (addrspacecast strips the aperture)
static __device__ __forceinline__ unsigned ldsOffset(const void* p) {
    return (unsigned)(unsigned long long)(__attribute__((address_space(3))) const void*)p;
}

// h = relu(edge_attr @ w1 + b1)   [E,5] x [5,128] -> [E,128]
__global__ void edge_mlp_kernel(const float* __restrict__ ea,
                                const float* __restrict__ w1,
                                const float* __restrict__ b1,
                                float* __restrict__ h, int E) {
    int idx = blockIdx.x * blockDim.x + threadIdx.x;
    if (idx >= E * 128) return;
    int e = idx >> 7, j = idx & 127;
    float s = b1[j];
    const float* er = ea + e * 5;
#pragma unroll
    for (int c = 0; c < 5; ++c) s += er[c] * w1[c * 128 + j];
    h[idx] = s > 0.0f ? s : 0.0f;
}

// xs[e,i] = x[src[e], i]  (zero-padded up to mi_pad)
__global__ void gather_kernel(const float* __restrict__ x,
                              const int* __restrict__ src,
                              float* __restrict__ xs,
                              int E, int mi_real, int mi_shift) {
    int idx = blockIdx.x * blockDim.x + threadIdx.x;
    if (idx >= (E << mi_shift)) return;
    int e = idx >> mi_shift;
    int i = idx & ((1 << mi_shift) - 1);
    xs[idx] = (i < mi_real) ? x[src[e] * mi_real + i] : 0.0f;
}

// Zero-pad w2/b2 of layer 0 from [128, 13*mo] to [128*16, mo] (+ [16, mo]).
__global__ void pack_w2_kernel(const float* __restrict__ w2,
                               const float* __restrict__ b2,
                               float* __restrict__ Vp, float* __restrict__ Bp,
                               int mi_real, int mi_pad, int mo_shift) {
    int idx = blockIdx.x * blockDim.x + threadIdx.x;
    const int mo = 1 << mo_shift;
    const int vtot = 128 * mi_pad * mo;
    if (idx < vtot) {
        int n = idx & (mo - 1);
        int t = idx >> mo_shift;            // k*mi_pad + i
        int i = t & (mi_pad - 1);
        int k = t >> 4;                     // mi_pad == 16 for the packed layer
        Vp[idx] = (i < mi_real) ? w2[(k * mi_real + i) * mo + n] : 0.0f;
    } else if (idx < vtot + (mi_pad << mo_shift)) {
        int r = idx - vtot;
        int n = r & (mo - 1);
        int i = r >> mo_shift;
        Bp[r] = (i < mi_real) ? b2[i * mo + n] : 0.0f;
    }
}

// Fused per-edge bilinear GEMM via fp32 WMMA, TDM-staged LDS tiles.
// Block: 64 edges (4 M-tiles of 16).  One wave per 16-wide N-tile.
// V is [128*MI_PAD, MO] dense (pre-packed for layer 0), Bp is [MI_PAD, MO].
template <int MI_PAD, int MO>
__global__ __launch_bounds__(32 * (MO / 16)) void
msg_wmma_kernel(const float* __restrict__ h,    // [E,128]
                const float* __restrict__ xs,   // [E,MI_PAD]
                const float* __restrict__ V,    // [128*MI_PAD, MO]
                const float* __restrict__ Bp,   // [MI_PAD, MO]
                float* __restrict__ msg,        // [E,MO]
                int E) {
    constexpr int KPI = MI_PAD / 4;        // 4-wide K-chunks per h-column
    constexpr int XLD = MI_PAD + 1;        // padded LDS row strides
    constexpr int XPADC = (MI_PAD == 16) ? 3 : (MI_PAD == 32) ? 4 : 5;

    __shared__ float hs[64 * 129];
    __shared__ float xss[64 * XLD];

    const int eBase = blockIdx.x * 64;
    int rows = E - eBase;
    if (rows > 64) rows = 64;

    if (threadIdx.x < 32) {
        // --- TDM: h tile [rows x 128] -> hs with 1-dword pad per 128 dwords ---
        unsigned long long hg = (unsigned long long)(const void*)(h + (size_t)eBase * 128);
        v4u g0;
        g0[0] = 1u;                                   // count=1, user mode
        g0[1] = ldsOffset(hs);                        // lds_addr
        g0[2] = (unsigned)hg;                         // global_addr[31:0]
        g0[3] = (unsigned)((hg >> 32) & 0x01FFFFFFull) | 0x80000000u;  // addr[56:32] | type=2
        v8u g1;
        g1[0] = (2u << 16) | (1u << 20) | (6u << 22); // data_size=4B, pad_en, every 128 dw
        g1[1] = (128u << 16);                         // tensor_dim0 = 128
        g1[2] = ((unsigned)rows << 16);               // tensor_dim1 = rows (OOB rows -> 0)
        g1[3] = (128u << 16);                         // tile_dim0 = 128
        g1[4] = 64u;                                  // tile_dim1 = 64
        g1[5] = 128u;                                 // tensor_dim0_stride = 128
        g1[6] = 0u;
        g1[7] = 0u;
        asm volatile("tensor_load_to_lds %0, %1" ::"s"(g0), "s"(g1) : "memory");

        // --- TDM: xs tile [rows x MI_PAD] -> xss with pad per MI_PAD dwords ---
        unsigned long long xg = (unsigned long long)(const void*)(xs + (size_t)eBase * MI_PAD);
        v4u g0x;
        g0x[0] = 1u;
        g0x[1] = ldsOffset(xss);
        g0x[2] = (unsigned)xg;
        g0x[3] = (unsigned)((xg >> 32) & 0x01FFFFFFull) | 0x80000000u;
        v8u g1x;
        g1x[0] = (2u << 16) | (1u << 20) | ((unsigned)XPADC << 22);
        g1x[1] = ((unsigned)MI_PAD << 16);
        g1x[2] = ((unsigned)rows << 16);
        g1x[3] = ((unsigned)MI_PAD << 16);
        g1x[4] = 64u;
        g1x[5] = (unsigned)MI_PAD;
        g1x[6] = 0u;
        g1x[7] = 0u;
        asm volatile("tensor_load_to_lds %0, %1" ::"s"(g0x), "s"(g1x) : "memory");

        __builtin_amdgcn_s_wait_tensorcnt(0);
    }
    __syncthreads();

    const int lane = threadIdx.x & 31;
    const int wv = threadIdx.x >> 5;     // N-tile index
    const int ml = lane & 15;
    const bool hi = lane >= 16;
    const int n = wv * 16 + ml;          // output column
    const int iOff = hi ? 2 : 0;         // A 16x4 f32: lanes16-31 hold K+2,K+3

    v8f acc[4] = {};

    for (int k = 0; k < 128; ++k) {
        float hv[4];
#pragma unroll
        for (int mt = 0; mt < 4; ++mt) hv[mt] = hs[(mt * 16 + ml) * 129 + k];
        const float* vk = V + (size_t)k * (MI_PAD * MO);
#pragma unroll
        for (int ci = 0; ci < KPI; ++ci) {
            const int iA = ci * 4 + iOff;
            v2f bf;
            bf[0] = vk[iA * MO + n];
            bf[1] = vk[(iA + 1) * MO + n];
#pragma unroll
            for (int mt = 0; mt < 4; ++mt) {
                const float* xr = &xss[(mt * 16 + ml) * XLD + iA];
                v2f af;
                af[0] = hv[mt] * xr[0];
                af[1] = hv[mt] * xr[1];
                acc[mt] = __builtin_amdgcn_wmma_f32_16x16x4_f32(
                    false, af, false, bf, (short)0, acc[mt], false, false);
            }
        }
    }

    // b2 contribution: msg += xs @ Bp   (A = xs, h-factor = 1)
#pragma unroll
    for (int ci = 0; ci < KPI; ++ci) {
        const int iA = ci * 4 + iOff;
        v2f bf;
        bf[0] = Bp[iA * MO + n];
        bf[1] = Bp[(iA + 1) * MO + n];
#pragma unroll
        for (int mt = 0; mt < 4; ++mt) {
            const float* xr = &xss[(mt * 16 + ml) * XLD + iA];
            v2f af;
            af[0] = xr[0];
            af[1] = xr[1];
            acc[mt] = __builtin_amdgcn_wmma_f32_16x16x4_f32(
                false, af, false, bf, (short)0, acc[mt], false, false);
        }
    }

    // C layout: VGPR v -> row v (lanes 0-15) / v+8 (lanes 16-31), col = lane&15.
#pragma unroll
    for (int mt = 0; mt < 4; ++mt) {
#pragma unroll
        for (int v = 0; v < 8; ++v) {
            int e = eBase + mt * 16 + v + (hi ? 8 : 0);
            if (e < E) msg[e * MO + n] = acc[mt][v];
        }
    }
}

// y = x @ root + bias   (pre-activation node term; overwrites y)
__global__ void root_bias_kernel(const float* __restrict__ x,
                                 const float* __restrict__ root,
                                 const float* __restrict__ bias,
                                 float* __restrict__ y,
                                 int N, int mi, int mo, int mo_shift) {
    int idx = blockIdx.x * blockDim.x + threadIdx.x;
    if (idx >= (N << mo_shift)) return;
    int nn = idx >> mo_shift;
    int o = idx & ((1 << mo_shift) - 1);
    float s = bias[o];
    const float* xr = x + nn * mi;
    for (int i = 0; i < mi; ++i) s += xr[i] * root[i * mo + o];
    y[idx] = s;
}

// y[dst[e], o] += msg[e, o]   (segment_sum via float atomics)
__global__ void scatter_kernel(const float* __restrict__ msg,
                               const int* __restrict__ dst,
                               float* __restrict__ y, int E, int mo_shift) {
    int idx = blockIdx.x * blockDim.x + threadIdx.x;
    if (idx >= (E << mo_shift)) return;
    int e = idx >> mo_shift;
    int o = idx & ((1 << mo_shift) - 1);
    atomicAdd(&y[(dst[e] << mo_shift) + o], msg[idx]);
}

__global__ void elu_kernel(float* __restrict__ y, int count) {
    int idx = blockIdx.x * blockDim.x + threadIdx.x;
    if (idx < count) y[idx] = eluf(y[idx]);
}

__global__ void zero_kernel(float* __restrict__ p, int count) {
    int idx = blockIdx.x * blockDim.x + threadIdx.x;
    if (idx < count) p[idx] = 0.0f;
}

// graph pooling sums + counts
__global__ void pool_kernel(const float* __restrict__ x,
                            const int* __restrict__ batch,
                            float* __restrict__ gsum,
                            float* __restrict__ gcnt, int N) {
    int idx = blockIdx.x * blockDim.x + threadIdx.x;
    if (idx >= N * 64) return;
    int nn = idx >> 6, o = idx & 63;
    int g = batch[nn];
    atomicAdd(&gsum[g * 64 + o], x[idx]);
    if (o == 0) atomicAdd(&gcnt[g], 1.0f);
}

// mean + fc1(elu) + fc2(elu) + fc3  (one thread per graph; tiny)
__global__ void head_kernel(const float* __restrict__ gsum,
                            const float* __restrict__ gcnt,
                            const float* __restrict__ fc1w, const float* __restrict__ fc1b,
                            const float* __restrict__ fc2w, const float* __restrict__ fc2b,
                            const float* __restrict__ fc3w, const float* __restrict__ fc3b,
                            float* __restrict__ out, int G) {
    int g = blockIdx.x * blockDim.x + threadIdx.x;
    if (g >= G) return;
    float inv = 1.0f / fmaxf(gcnt[g], 1.0f);
    float xg[64];
    for (int i = 0; i < 64; ++i) xg[i] = gsum[g * 64 + i] * inv;
    float h1[32];
    for (int j = 0; j < 32; ++j) {
        float s = fc1b[j];
        for (int i = 0; i < 64; ++i) s += xg[i] * fc1w[i * 32 + j];
        h1[j] = eluf(s);
    }
    float h2[16];
    for (int j = 0; j < 16; ++j) {
        float s = fc2b[j];
        for (int i = 0; i < 32; ++i) s += h1[i] * fc2w[i * 16 + j];
        h2[j] = eluf(s);
    }
    float s = fc3b[0];
    for (int i = 0; i < 16; ++i) s += h2[i] * fc3w[i];
    out[g] = s;
}

extern "C" void kernel_launch(void* const* d_in, const int* in_sizes, int n_in,
                              void* d_out, int out_size, void* d_ws, size_t ws_size,
                              hipStream_t stream) {
    (void)n_in; (void)ws_size;
    const float* x_in  = (const float*)d_in[0];
    const int*   ei    = (const int*)d_in[1];
    const float* ea    = (const float*)d_in[2];
    const int*   batch = (const int*)d_in[3];
    const int E = in_sizes[1] / 2;
    const int N = in_sizes[3];
    const int G = out_size;
    const int* src = ei;
    const int* dst = ei + E;

    float* ws = (float*)d_ws;
    float* h    = ws; ws += (size_t)E * 128;
    float* xs   = ws; ws += (size_t)E * 64;
    float* msg  = ws; ws += (size_t)E * 64;
    float* xa   = ws; ws += (size_t)N * 64;
    float* xb   = ws; ws += (size_t)N * 64;
    float* gsum = ws; ws += (size_t)G * 64;
    float* gcnt = ws; ws += G;
    float* vp   = ws; ws += (size_t)128 * 16 * 32;   // packed w2_0
    float* bp   = ws;                                // packed b2_0
    float* out = (float*)d_out;

    struct Layer {
        const float *w1, *b1, *w2, *b2, *root, *bias;
        int mi_real, mi_pad, mi_shift, mo, mo_shift;
    } L[3];
    const int MIR[3] = {13, 32, 64}, MIP[3] = {16, 32, 64}, MIS[3] = {4, 5, 6};
    const int MOV[3] = {32, 64, 64}, MOS[3] = {5, 6, 6};
    for (int l = 0; l < 3; ++l) {
        int b = 4 + l * 6;
        L[l].w1 = (const float*)d_in[b + 0];
        L[l].b1 = (const float*)d_in[b + 1];
        L[l].w2 = (const float*)d_in[b + 2];
        L[l].b2 = (const float*)d_in[b + 3];
        L[l].root = (const float*)d_in[b + 4];
        L[l].bias = (const float*)d_in[b + 5];
        L[l].mi_real = MIR[l]; L[l].mi_pad = MIP[l]; L[l].mi_shift = MIS[l];
        L[l].mo = MOV[l]; L[l].mo_shift = MOS[l];
    }

    // Pre-pack layer 0's w2/b2 to padded K=16 layout (layers 1/2 are already dense).
    {
        int tot = 128 * 16 * 32 + 16 * 32;
        pack_w2_kernel<<<(tot + 255) / 256, 256, 0, stream>>>(
            L[0].w2, L[0].b2, vp, bp, 13, 16, 5);
    }

    const float* xcur = x_in;
    float* bufs[2] = {xa, xb};
    for (int l = 0; l < 3; ++l) {
        const Layer& P = L[l];
        int nb = (E * 128 + 255) / 256;
        edge_mlp_kernel<<<nb, 256, 0, stream>>>(ea, P.w1, P.b1, h, E);

        nb = ((E << P.mi_shift) + 255) / 256;
        gather_kernel<<<nb, 256, 0, stream>>>(xcur, src, xs, E, P.mi_real, P.mi_shift);

        int nblk = (E + 63) / 64;
        if (l == 0) {
            msg_wmma_kernel<16, 32><<<nblk, 64, 0, stream>>>(h, xs, vp, bp, msg, E);
        } else if (l == 1) {
            msg_wmma_kernel<32, 64><<<nblk, 128, 0, stream>>>(h, xs, P.w2, P.b2, msg, E);
        } else {
            msg_wmma_kernel<64, 64><<<nblk, 128, 0, stream>>>(h, xs, P.w2, P.b2, msg, E);
        }

        float* xnext = bufs[l & 1];   // l0->xa, l1->xb, l2->xa (always != xcur)
        nb = ((N << P.mo_shift) + 255) / 256;
        root_bias_kernel<<<nb, 256, 0, stream>>>(xcur, P.root, P.bias, xnext,
                                                 N, P.mi_real, P.mo, P.mo_shift);
        nb = ((E << P.mo_shift) + 255) / 256;
        scatter_kernel<<<nb, 256, 0, stream>>>(msg, dst, xnext, E, P.mo_shift);
        nb = ((N << P.mo_shift) + 255) / 256;
        elu_kernel<<<nb, 256, 0, stream>>>(xnext, N << P.mo_shift);
        xcur = xnext;
    }

    // gsum and gcnt are contiguous in ws: zero both in one launch.
    zero_kernel<<<(G * 64 + G + 255) / 256, 256, 0, stream>>>(gsum, G * 64 + G);
    pool_kernel<<<(N * 64 + 255) / 256, 256, 0, stream>>>(xcur, batch, gsum, gcnt, N);
    head_kernel<<<(G + 127) / 128, 128, 0, stream>>>(
        gsum, gcnt,
        (const float*)d_in[22], (const float*)d_in[23],
        (const float*)d_in[24], (const float*)d_in[25],
        (const float*)d_in[26], (const float*)d_in[27],
        out, G);
}